// ProbabilisticChamferDistanceLoss_15925738734020
// MI455X (gfx1250) — compile-verified
//
#include <hip/hip_runtime.h>

#define N_ORIG 30000
#define M_SIMP 10000
#define RTILES (N_ORIG / 16)   // 1875 row tiles (exact)
#define CTILES (M_SIMP / 16)   // 625 col tiles (exact)
#define WAVES_PER_BLOCK 8
#define BLOCK_THREADS (WAVES_PER_BLOCK * 32)
#define GRID_MAIN ((RTILES + WAVES_PER_BLOCK - 1) / WAVES_PER_BLOCK) // 235

typedef __attribute__((ext_vector_type(2))) float v2f;
typedef __attribute__((ext_vector_type(8))) float v8f;

// ---------------------------------------------------------------- init
__global__ void pcd_init_ws(unsigned* __restrict__ colmin) {
    int j = blockIdx.x * blockDim.x + threadIdx.x;
    if (j < M_SIMP) colmin[j] = 0x7F800000u;   // +inf bits
}

// ---------------------------------------------------------------- main
__global__ __launch_bounds__(BLOCK_THREADS)
void pcd_main(const float* __restrict__ P, const float* __restrict__ Ps,
              const float* __restrict__ prob,
              unsigned* __restrict__ gcolmin, float* __restrict__ rowpart) {
    __shared__ unsigned lcol[M_SIMP];          // 40 KB of the 320 KB WGP LDS
    const int tid = threadIdx.x;
    for (int j = tid; j < M_SIMP; j += BLOCK_THREADS) lcol[j] = 0x7F800000u;
    __syncthreads();

    const int lane = tid & 31;
    const int wave = tid >> 5;
    const int rowtile = blockIdx.x * WAVES_PER_BLOCK + wave;   // wave-uniform
    if (rowtile < RTILES) {
        const int l15  = lane & 15;
        const int half = lane >> 4;
        const int rowbase = rowtile * 16;

        // A matrix (16x4 f32): lanes 0-15 hold (x,y) K=0,1; lanes 16-31 hold (z,0) K=2,3.
        // Pre-scaled by -2 so the WMMA yields -2*P.Ps directly.
        const float* pa = P + 3 * (rowbase + l15);
        const float ax = pa[0], ay = pa[1], az = pa[2];
        v2f a;
        a.x = -2.0f * (half ? az : ax);
        a.y = half ? 0.0f : (-2.0f * ay);

        // |P|^2 for the 8 rows this lane's half owns in the C/D layout (M = v + 8*half)
        float p2v[8];
        #pragma unroll
        for (int v = 0; v < 8; ++v) {
            const float* pr = P + 3 * (rowbase + v + 8 * half);
            p2v[v] = pr[0] * pr[0] + pr[1] * pr[1] + pr[2] * pr[2];
        }

        float rv[8];                 // running min of d^2 per owned row
        int   ri[8];
        #pragma unroll
        for (int v = 0; v < 8; ++v) { rv[v] = __builtin_inff(); ri[v] = 0; }

        // software-pipelined Ps load: fetch tile ct+1 while computing tile ct
        const float* pb0 = Ps + 3 * l15;
        float cbx = pb0[0], cby = pb0[1], cbz = pb0[2];

        for (int ct = 0; ct < CTILES; ++ct) {
            const int ncol = ((ct + 1 < CTILES) ? (ct + 1) * 16 : 0) + l15;
            const float* pn = Ps + 3 * ncol;
            const float nbx = pn[0], nby = pn[1], nbz = pn[2];  // prefetch next

            const int col = ct * 16 + l15;       // this lane's column (N = l15 in C/D)
            const float q2 = cbx * cbx + cby * cby + cbz * cbz;
            v2f b;                               // B (4x16): mirror of A layout
            b.x = half ? cbz : cbx;
            b.y = half ? 0.0f : cby;

            v8f c = {};
            v8f acc = __builtin_amdgcn_wmma_f32_16x16x4_f32(
                false, a, false, b, (short)0, c, false, false);

            float cmin = __builtin_inff();
            #pragma unroll
            for (int v = 0; v < 8; ++v) {
                float d2 = (p2v[v] + q2) + acc[v];       // p2 + q2 - 2*dot
                if (d2 < rv[v]) { rv[v] = d2; ri[v] = col; }
                cmin = fminf(cmin, d2);
            }
            // combine the two 8-row halves of this column; clamp so u32 order == float order
            cmin = fminf(cmin, __shfl_xor(cmin, 16, 32));
            cmin = fmaxf(cmin, 0.0f);
            if (half == 0) atomicMin(&lcol[col], __float_as_uint(cmin));

            cbx = nbx; cby = nby; cbz = nbz;     // rotate pipeline
        }

        // row min-d^2 + argmin across the 16 lanes of each half-group
        #pragma unroll
        for (int v = 0; v < 8; ++v) {
            #pragma unroll
            for (int m = 8; m >= 1; m >>= 1) {
                float ov = __shfl_xor(rv[v], m, 32);
                int   oi = __shfl_xor(ri[v], m, 32);
                if (ov < rv[v] || (ov == rv[v] && oi < ri[v])) { rv[v] = ov; ri[v] = oi; }
            }
        }
        float rsum = 0.0f;
        if (l15 == 0) {                       // lane 0 -> rows 0..7, lane 16 -> rows 8..15
            #pragma unroll
            for (int v = 0; v < 8; ++v) {
                float d = __builtin_amdgcn_sqrtf(fmaxf(rv[v], 0.0f));  // raw v_sqrt_f32
                rsum += d * prob[ri[v]];
            }
        }
        rsum += __shfl_xor(rsum, 16, 32);
        if (lane == 0) rowpart[rowtile] = rsum;   // deterministic per-tile partial
    }
    __syncthreads();
    // merge block-local column min-d^2 into global (order-independent u32 min)
    for (int j = tid; j < M_SIMP; j += BLOCK_THREADS)
        atomicMin(&gcolmin[j], lcol[j]);
}

// ---------------------------------------------------------------- finish
__global__ __launch_bounds__(256)
void pcd_finish(const unsigned* __restrict__ colmin, const float* __restrict__ prob,
                const float* __restrict__ rowpart, float* __restrict__ out) {
    __shared__ float red[256];
    float s = 0.0f;
    for (int j = threadIdx.x; j < M_SIMP; j += 256)
        s += __builtin_amdgcn_sqrtf(__uint_as_float(colmin[j])) * prob[j];
    for (int t = threadIdx.x; t < RTILES; t += 256)
        s += rowpart[t];                               // sum(dist_o_to_s * prob[argmin])
    red[threadIdx.x] = s;
    __syncthreads();
    for (int w = 128; w > 0; w >>= 1) {
        if (threadIdx.x < w) red[threadIdx.x] += red[threadIdx.x + w];
        __syncthreads();
    }
    if (threadIdx.x == 0) out[0] = red[0];
}

// ---------------------------------------------------------------- launch
extern "C" void kernel_launch(void* const* d_in, const int* in_sizes, int n_in,
                              void* d_out, int out_size, void* d_ws, size_t ws_size,
                              hipStream_t stream) {
    const float* P    = (const float*)d_in[0];   // (30000, 3)
    const float* Ps   = (const float*)d_in[1];   // (10000, 3)
    const float* prob = (const float*)d_in[2];   // (10000,)

    unsigned* colmin  = (unsigned*)d_ws;                                   // M u32
    float*    rowpart = (float*)((char*)d_ws + M_SIMP * sizeof(unsigned)); // RTILES f32
    float*    out     = (float*)d_out;

    hipLaunchKernelGGL(pcd_init_ws, dim3((M_SIMP + 255) / 256), dim3(256), 0, stream,
                       colmin);
    hipLaunchKernelGGL(pcd_main, dim3(GRID_MAIN), dim3(BLOCK_THREADS), 0, stream,
                       P, Ps, prob, colmin, rowpart);
    hipLaunchKernelGGL(pcd_finish, dim3(1), dim3(256), 0, stream,
                       colmin, prob, rowpart, out);
}